// kernel_generated_0_62904091017484
// MI455X (gfx1250) — compile-verified
//
#include <hip/hip_runtime.h>
#include <hip/hip_bf16.h>

// ---------------------------------------------------------------------------
// Shapes (compile-time constants from the reference)
// ---------------------------------------------------------------------------
#define LDIM   128
#define JDIM   24          // input channels (j); output channels (k) = 24 too
#define MDIM   112
#define NDIM   112
#define KOUT   24
#define QDIM   240         // GEMM K: 24 j-channels * 10 merged taps
#define KPAD   32          // k padded to two 16x16 WMMA row tiles
#define XSTR   132         // LDS stride for one x row (112 + halo, bank-staggered)
#define GPSTR  68          // LDS stride per q-pair row (64 payload + 4 pad)
#define ROWS   2           // (l,m) rows per workgroup

typedef __attribute__((ext_vector_type(2))) float v2f;
typedef __attribute__((ext_vector_type(8))) float v8f;

// ---------------------------------------------------------------------------
// Prep kernel: build c0,c1 (k2-split contracted weights) and gP (merged
// 10-tap filter, pair-interleaved for b64 WMMA A-fragment loads).
//   c_{k2}[k,j,kh] = sum_i w1[k,kh,i] * w2[j,k2,i]
//   g[k, q=j*10+dt] = (d>=-4 ? c0[k,j,d+4] : 0) + (d<=3 ? c1[k,j,d+5] : 0),
//                     d = dt-5
//   gP[(q>>1)*64 + k*2 + (q&1)] = g[k,q]   (k in [0,32), zero-padded)
// ---------------------------------------------------------------------------
__global__ void prep_weights(const float* __restrict__ w1,
                             const float* __restrict__ w2,
                             float* __restrict__ gP,
                             float* __restrict__ c0,
                             float* __restrict__ c1) {
  int idx = blockIdx.x * blockDim.x + threadIdx.x;
  if (idx < 2 * 24 * 24 * 9) {                       // c part: 10368 entries
    int k2 = idx / 5184;
    int r  = idx - k2 * 5184;
    int k  = r / 216;
    int r2 = r - k * 216;
    int j  = r2 / 9;
    int kh = r2 - j * 9;
    float s = 0.f;
    #pragma unroll
    for (int i = 0; i < 14; ++i)
      s += w1[k * 126 + kh * 14 + i] * w2[j * 28 + k2 * 14 + i];
    (k2 ? c1 : c0)[(k * 24 + j) * 9 + kh] = s;
  } else if (idx < 2 * 24 * 24 * 9 + QDIM * KPAD) {  // g part: 7680 entries
    int t  = idx - 2 * 24 * 24 * 9;
    int q  = t >> 5;
    int k  = t & 31;
    int j  = q / 10;
    int dt = q - j * 10;
    int d  = dt - 5;
    float s = 0.f;
    if (k < KOUT) {
      if (d >= -4) {
        int kh = d + 4;
        #pragma unroll
        for (int i = 0; i < 14; ++i)
          s += w1[k * 126 + kh * 14 + i] * w2[j * 28 + i];          // k2 = 0
      }
      if (d <= 3) {
        int kh = d + 5;
        #pragma unroll
        for (int i = 0; i < 14; ++i)
          s += w1[k * 126 + kh * 14 + i] * w2[j * 28 + 14 + i];     // k2 = 1
      }
    }
    gP[(q >> 1) * 64 + k * 2 + (q & 1)] = s;
  }
}

// ---------------------------------------------------------------------------
// Main kernel: per (l,m) row, D[24x112] = G[24x240] * X_im2col[240x112]
// via V_WMMA_F32_16X16X4_F32. 4 waves/block: wave = (row in block, k-tile).
// Each wave keeps 7 n-tile accumulators (56 VGPRs) across the fully unrolled
// 60-step K loop; all (j,tap) indexing constant-folds at compile time.
// ---------------------------------------------------------------------------
__global__ __launch_bounds__(128)
void conv_wmma(const float* __restrict__ x,
               const float* __restrict__ gP,
               float* __restrict__ out) {
  __shared__ __align__(16) float xs[ROWS * JDIM * XSTR];     // 6336 f
  __shared__ __align__(16) float gs[(QDIM / 2) * GPSTR];     // 8160 f (~58 KB total)

  const int tid  = threadIdx.x;
  const int row0 = blockIdx.x * ROWS;

  // Stage weights (float4 global reads, padded q-pair rows in LDS)
  for (int t = tid; t < (QDIM / 2) * 64 / 4; t += 128) {
    int w  = t * 4;
    int qp = w >> 6, off = w & 63;
    *(float4*)&gs[qp * GPSTR + off] = *(const float4*)&gP[w];
  }
  // Stage x rows with zero halo: position n stored at p = n + 8, p in [0,128)
  for (int t = tid; t < ROWS * JDIM * 32; t += 128) {
    int p4 = t & 31;            // float4 slot, p = p4*4
    int rj = t >> 5;
    int r  = rj / JDIM;
    int j  = rj - r * JDIM;
    int rg = row0 + r;
    int l  = rg / MDIM;
    int m  = rg - l * MDIM;
    int p  = p4 * 4;
    float4 v = make_float4(0.f, 0.f, 0.f, 0.f);
    if (p >= 8 && p < 120)      // n = p-8 in [0,112), 16B-aligned in x
      v = *(const float4*)&x[(((size_t)l * JDIM + j) * MDIM + m) * (size_t)NDIM + (p - 8)];
    *(float4*)&xs[(r * JDIM + j) * XSTR + p] = v;
  }
  __syncthreads();

  const int lane  = tid & 31;
  const int wave  = tid >> 5;
  const int half  = lane >> 4;      // lane half selects K pair / C row block
  const int nidx  = lane & 15;
  const int rsel  = wave >> 1;      // row within block
  const int ktile = wave & 1;
  // A-fragment base: addr(kb) = (kb/2)*GPSTR + abase  (kb/2 part is a literal)
  const int abase = half * GPSTR + (ktile * 16 + nidx) * 2;
  // B-fragment base: addr = bbase + sel(half, per-kb consts) + t*16
  const int bbase = rsel * (JDIM * XSTR) + 3 + nidx;   // +8 halo, -5 tap base

  v8f acc[7];
  #pragma unroll
  for (int t = 0; t < 7; ++t) acc[t] = v8f{0.f,0.f,0.f,0.f,0.f,0.f,0.f,0.f};

  #pragma unroll
  for (int kb = 0; kb < QDIM; kb += 4) {
    // A: one b64 LDS read (pair-interleaved layout, 8B aligned)
    v2f a = *(const v2f*)&gs[(kb >> 1) * GPSTR + abase];
    // B row offsets: q -> (j = q/10, tap dt = q%10); all literals after unroll
    const int o0a = ((kb    ) / 10) * XSTR + ((kb    ) % 10);
    const int o1a = ((kb + 1) / 10) * XSTR + ((kb + 1) % 10);
    const int o0b = ((kb + 2) / 10) * XSTR + ((kb + 2) % 10);
    const int o1b = ((kb + 3) / 10) * XSTR + ((kb + 3) % 10);
    const int off0 = bbase + (half ? o0b : o0a);
    const int off1 = bbase + (half ? o1b : o1a);
    #pragma unroll
    for (int t = 0; t < 7; ++t) {
      v2f b;
      b.x = xs[off0 + t * 16];
      b.y = xs[off1 + t * 16];
      acc[t] = __builtin_amdgcn_wmma_f32_16x16x4_f32(
          /*neg_a=*/false, a, /*neg_b=*/false, b,
          /*c_mod=*/(short)0, acc[t], /*reuse_a=*/false, /*reuse_b=*/false);
    }
  }

  // Store: C/D layout -> element (k = ktile*16 + 8*half + r, n = t*16 + nidx)
  const int rg = row0 + rsel;
  const int l  = rg / MDIM;
  const int m  = rg - l * MDIM;
  #pragma unroll
  for (int t = 0; t < 7; ++t) {
    int n = t * 16 + nidx;
    #pragma unroll
    for (int r = 0; r < 8; ++r) {
      int k = ktile * 16 + half * 8 + r;
      if (k < KOUT)
        out[(((size_t)l * KOUT + k) * MDIM + m) * (size_t)NDIM + n] = acc[t][r];
    }
  }
}

// ---------------------------------------------------------------------------
// Fixup: column n=0 wraps through the rolled axis (uses x[...,107..111]).
// Overwrite out[:, :, :, 0] with the exact value.
// ---------------------------------------------------------------------------
__global__ void fixup_col0(const float* __restrict__ x,
                           const float* __restrict__ c0,
                           const float* __restrict__ c1,
                           float* __restrict__ out) {
  int tid = blockIdx.x * blockDim.x + threadIdx.x;
  if (tid >= LDIM * KOUT * MDIM) return;
  int l = tid / (KOUT * MDIM);
  int r = tid - l * KOUT * MDIM;
  int k = r / MDIM;
  int m = r - k * MDIM;
  float s = 0.f;
  for (int j = 0; j < JDIM; ++j) {
    const float* xr = x + (((size_t)l * JDIM + j) * MDIM + m) * (size_t)NDIM;
    const float* C0 = c0 + (k * 24 + j) * 9;
    const float* C1 = c1 + (k * 24 + j) * 9;
    #pragma unroll
    for (int kh = 4; kh < 9; ++kh) s += C0[kh] * xr[kh - 4];      // k2=0 taps
    #pragma unroll
    for (int kh = 0; kh < 5; ++kh) s += C1[kh] * xr[107 + kh];    // k2=1 wrap
  }
  out[(((size_t)l * KOUT + k) * MDIM + m) * (size_t)NDIM] = s;
}

// ---------------------------------------------------------------------------
extern "C" void kernel_launch(void* const* d_in, const int* in_sizes, int n_in,
                              void* d_out, int out_size, void* d_ws, size_t ws_size,
                              hipStream_t stream) {
  const float* x  = (const float*)d_in[0];   // [128,24,112,112]
  const float* w1 = (const float*)d_in[1];   // [24,9,14]
  const float* w2 = (const float*)d_in[2];   // [24,2,14]
  float* outf = (float*)d_out;               // [128,24,112,112]

  // Workspace layout (72 KB total): gP[120*64] | c0[24*24*9] | c1[24*24*9]
  float* wsf = (float*)d_ws;
  float* gP = wsf;
  float* c0 = wsf + QDIM * KPAD;
  float* c1 = c0 + 24 * 24 * 9;

  int prep_n = 2 * 24 * 24 * 9 + QDIM * KPAD;                 // 18048
  prep_weights<<<(prep_n + 255) / 256, 256, 0, stream>>>(w1, w2, gP, c0, c1);

  conv_wmma<<<(LDIM * MDIM) / ROWS, 128, 0, stream>>>(x, gP, outf);

  int fix_n = LDIM * KOUT * MDIM;                             // 344064
  fixup_col0<<<(fix_n + 255) / 256, 256, 0, stream>>>(x, c0, c1, outf);
}